// BasicResidualBlock_67276367725113
// MI455X (gfx1250) — compile-verified
//
#include <hip/hip_runtime.h>

typedef float v2f __attribute__((ext_vector_type(2)));
typedef float v8f __attribute__((ext_vector_type(8)));

constexpr int N    = 65536;
constexpr int C    = 64;
constexpr int K    = 27;
constexpr int ROWS = 128;           // gathered rows per workgroup
constexpr int MBLK = N / ROWS;      // 512 row-blocks
constexpr int GSTR = 68;            // LDS row stride in floats (bank-conflict pad)
constexpr float EPS = 1e-5f;

// ---------------------------------------------------------------- zero scratch
__global__ __launch_bounds__(256)
void zero_ws_kernel(float* __restrict__ ws, long n) {
    long i = (long)blockIdx.x * blockDim.x + threadIdx.x;
    long stride = (long)gridDim.x * blockDim.x;
    for (; i < n; i += stride) ws[i] = 0.0f;
}

// -------------------------------------------------- gather -> WMMA -> scatter
// One block handles kernel-tap k and 128 consecutive "m" rows.
// If mean != nullptr, relu(BN) is applied on the fly to gathered inputs.
__global__ __launch_bounds__(256)
void sconv_wmma_kernel(const float* __restrict__ src,      // [N,C] input feats
                       const float* __restrict__ W,        // [K,C,C]
                       const int*   __restrict__ in_idx,   // [K,N]
                       const int*   __restrict__ out_idx,  // [K,N]
                       const float* __restrict__ mean,     // [C] or nullptr
                       const float* __restrict__ scale,    // [C] gamma*rsqrt(var+eps)
                       const float* __restrict__ beta,     // [C]
                       float*       __restrict__ dst)      // [N,C] accumulator
{
    __shared__ __align__(16) float sG[ROWS * GSTR];   // gathered rows (padded)
    __shared__ __align__(16) float sW[C * GSTR];      // W[k]^T : sW[co*GSTR+ci]
    __shared__ int   sIn[ROWS], sOut[ROWS];
    __shared__ float sMean[C], sScale[C], sBeta[C];

    const int t  = threadIdx.x;
    const int k  = blockIdx.x / MBLK;
    const int m0 = (blockIdx.x % MBLK) * ROWS;
    const bool bn = (mean != nullptr);

    if (t < ROWS) {
        sIn[t]  = in_idx [k * N + m0 + t];
        sOut[t] = out_idx[k * N + m0 + t];
    }
    if (bn && t < C) {
        sMean[t]  = mean[t];
        sScale[t] = scale[t];
        sBeta[t]  = beta[t];
    }
    // stage W[k] transposed (memory: [c_in][c_out] -> LDS: [c_out][c_in])
    const float* Wk = W + (long)k * C * C;
    for (int idx = t; idx < C * C; idx += 256) {
        int ci = idx >> 6, co = idx & 63;
        sW[co * GSTR + ci] = Wk[idx];
    }
    __syncthreads();

    // gather 128 rows x 64ch as float4 chunks (2048 chunks / 256 threads)
    for (int i = 0; i < 8; ++i) {
        int linear = t + 256 * i;
        int row = linear >> 4;
        int q   = linear & 15;
        long grow = sIn[row];
        float4 v = ((const float4*)(src + grow * C))[q];
        if (bn) {
            int c = q * 4;
            v.x = fmaxf(0.0f, (v.x - sMean[c + 0]) * sScale[c + 0] + sBeta[c + 0]);
            v.y = fmaxf(0.0f, (v.y - sMean[c + 1]) * sScale[c + 1] + sBeta[c + 1]);
            v.z = fmaxf(0.0f, (v.z - sMean[c + 2]) * sScale[c + 2] + sBeta[c + 2]);
            v.w = fmaxf(0.0f, (v.w - sMean[c + 3]) * sScale[c + 3] + sBeta[c + 3]);
        }
        *((float4*)&sG[row * GSTR + q * 4]) = v;   // (row*272 + q*16) bytes: 16B aligned
    }
    __syncthreads();

    // each wave computes a 16x64 output tile: D = A(16x64) * W(64x64)
    const int lane   = t & 31;
    const int w      = t >> 5;            // wave 0..7 -> rows 16w..16w+15
    const int laneLo = lane & 15;
    const int hi     = lane >> 4;         // 0: K,K+1  1: K+2,K+3 (f32 A/B layout)

    const float* gRow = &sG[(16 * w + laneLo) * GSTR];
    const float* bCol = &sW[laneLo * GSTR];

    v8f acc0 = {}, acc1 = {}, acc2 = {}, acc3 = {};
    for (int kk = 0; kk < 16; ++kk) {
        const int kc = 4 * kk + 2 * hi;
        v2f a  = *((const v2f*)&gRow[kc]);
        v2f b0 = *((const v2f*)&bCol[0 * 16 * GSTR + kc]);
        v2f b1 = *((const v2f*)&bCol[1 * 16 * GSTR + kc]);
        v2f b2 = *((const v2f*)&bCol[2 * 16 * GSTR + kc]);
        v2f b3 = *((const v2f*)&bCol[3 * 16 * GSTR + kc]);
        acc0 = __builtin_amdgcn_wmma_f32_16x16x4_f32(false, a, false, b0, (short)0, acc0, false, false);
        acc1 = __builtin_amdgcn_wmma_f32_16x16x4_f32(false, a, false, b1, (short)0, acc1, false, false);
        acc2 = __builtin_amdgcn_wmma_f32_16x16x4_f32(false, a, false, b2, (short)0, acc2, false, false);
        acc3 = __builtin_amdgcn_wmma_f32_16x16x4_f32(false, a, false, b3, (short)0, acc3, false, false);
    }

    // scatter-add: D vgpr r holds (M = r + 8*hi, Ncol = 16n + laneLo)
    for (int r = 0; r < 8; ++r) {
        int rowLocal = 16 * w + r + 8 * hi;
        long orow = sOut[rowLocal];
        float* outp = dst + orow * C + laneLo;
        atomicAdd(outp + 0,  acc0[r]);
        atomicAdd(outp + 16, acc1[r]);
        atomicAdd(outp + 32, acc2[r]);
        atomicAdd(outp + 48, acc3[r]);
    }
}

// ------------------------------------------------------------ BN statistics
__global__ __launch_bounds__(256)
void bn_stats_kernel(const float* __restrict__ y,
                     float* __restrict__ sum, float* __restrict__ sq) {
    __shared__ float sS[4][C], sQ[4][C];
    const int t  = threadIdx.x;
    const int ch = t & 63;
    const int g  = t >> 6;                 // 4 row groups of 256 rows
    long r0 = (long)blockIdx.x * 1024 + g * 256;
    float s = 0.0f, q = 0.0f;
    for (int i = 0; i < 256; ++i) {
        float v = y[(r0 + i) * C + ch];
        s += v;
        q += v * v;
    }
    sS[g][ch] = s;
    sQ[g][ch] = q;
    __syncthreads();
    if (t < C) {
        float ts = sS[0][t] + sS[1][t] + sS[2][t] + sS[3][t];
        float tq = sQ[0][t] + sQ[1][t] + sQ[2][t] + sQ[3][t];
        atomicAdd(&sum[t], ts);
        atomicAdd(&sq[t], tq);
    }
}

__global__ void bn_finalize_kernel(const float* __restrict__ sum,
                                   const float* __restrict__ sq,
                                   const float* __restrict__ gamma,
                                   float* __restrict__ mean,
                                   float* __restrict__ scale) {
    int c = threadIdx.x;
    if (c < C) {
        float m   = sum[c] * (1.0f / N);
        float var = sq[c] * (1.0f / N) - m * m;
        mean[c]  = m;
        scale[c] = gamma[c] * rsqrtf(var + EPS);
    }
}

// ---------------------------------------------- final BN + residual + relu
__global__ __launch_bounds__(256)
void bn_residual_relu_kernel(const float* __restrict__ y2,
                             const float* __restrict__ x,
                             const float* __restrict__ mean,
                             const float* __restrict__ scale,
                             const float* __restrict__ beta,
                             float* __restrict__ out) {
    long i = (long)blockIdx.x * blockDim.x + threadIdx.x;
    long base = i * 4;
    int c = (int)(base & 63);
    float4 v  = *((const float4*)(y2 + base));
    float4 xi = *((const float4*)(x + base));
    float4 o;
    o.x = fmaxf(0.0f, (v.x - mean[c + 0]) * scale[c + 0] + beta[c + 0] + xi.x);
    o.y = fmaxf(0.0f, (v.y - mean[c + 1]) * scale[c + 1] + beta[c + 1] + xi.y);
    o.z = fmaxf(0.0f, (v.z - mean[c + 2]) * scale[c + 2] + beta[c + 2] + xi.z);
    o.w = fmaxf(0.0f, (v.w - mean[c + 3]) * scale[c + 3] + beta[c + 3] + xi.w);
    *((float4*)(out + base)) = o;
}

// --------------------------------------------------------------------- launch
extern "C" void kernel_launch(void* const* d_in, const int* in_sizes, int n_in,
                              void* d_out, int out_size, void* d_ws, size_t ws_size,
                              hipStream_t stream) {
    const float* x      = (const float*)d_in[0];
    const float* W1     = (const float*)d_in[1];
    const float* gamma1 = (const float*)d_in[2];
    const float* beta1  = (const float*)d_in[3];
    const float* W2     = (const float*)d_in[4];
    const float* gamma2 = (const float*)d_in[5];
    const float* beta2  = (const float*)d_in[6];
    const int*   in_idx  = (const int*)d_in[7];
    const int*   out_idx = (const int*)d_in[8];
    float* out = (float*)d_out;

    float* ws = (float*)d_ws;
    float* y1 = ws;                       // [N,C]
    float* y2 = ws + (long)N * C;         // [N,C]
    float* st = ws + 2L * N * C;          // stats block
    float* sum1 = st +   0, *sq1 = st +  64, *mean1 = st + 128, *scale1 = st + 192;
    float* sum2 = st + 256, *sq2 = st + 320, *mean2 = st + 384, *scale2 = st + 448;

    // 1) zero y1, y2 and stats (graph-replay safe)
    long nz = 2L * N * C + 512;
    zero_ws_kernel<<<1024, 256, 0, stream>>>(ws, nz);

    // 2) conv1: y1 = scatter( gather(x) @ W1 )
    sconv_wmma_kernel<<<K * MBLK, 256, 0, stream>>>(
        x, W1, in_idx, out_idx, nullptr, nullptr, nullptr, y1);

    // 3) BN1 stats
    bn_stats_kernel<<<N / 1024, 256, 0, stream>>>(y1, sum1, sq1);
    bn_finalize_kernel<<<1, 64, 0, stream>>>(sum1, sq1, gamma1, mean1, scale1);

    // 4) conv2 with relu(BN1) fused into the gather: y2 = scatter( relu(bn(y1)) @ W2 )
    sconv_wmma_kernel<<<K * MBLK, 256, 0, stream>>>(
        y1, W2, in_idx, out_idx, mean1, scale1, beta1, y2);

    // 5) BN2 stats
    bn_stats_kernel<<<N / 1024, 256, 0, stream>>>(y2, sum2, sq2);
    bn_finalize_kernel<<<1, 64, 0, stream>>>(sum2, sq2, gamma2, mean2, scale2);

    // 6) out = relu(bn2(y2) + x)
    bn_residual_relu_kernel<<<(N * C / 4) / 256, 256, 0, stream>>>(
        y2, x, mean2, scale2, beta2, out);
}